// OutlierQuantLinear_82094004896527
// MI455X (gfx1250) — compile-verified
//
#include <hip/hip_runtime.h>
#include <hip/hip_bf16.h>

typedef __attribute__((ext_vector_type(2))) float v2f;
typedef __attribute__((ext_vector_type(8))) float v8f;

#define OUT_F 11008
#define IN_F  4096
#define BATCH 8
#define WAVES_PER_BLOCK 8
#define KSLICE (IN_F / WAVES_PER_BLOCK)   // 512 K-elements per wave

// y[b][o] = sum_i x[b][i] * (outl[o][i]*upd[o][i] + cw[o][i]) + bias[o]
// One block = one 16-wide output tile (N), 8 waves split K; WMMA f32 16x16x4.
__global__ __launch_bounds__(256, 2)
void OutlierQuantLinear_wmma_kernel(const float* __restrict__ x,
                                    const float* __restrict__ outliers,
                                    const float* __restrict__ updates,
                                    const float* __restrict__ cweight,
                                    const float* __restrict__ bias,
                                    float* __restrict__ out)
{
    // padded inner dim (9) -> lane stride 9 dwords, coprime with 64 banks
    __shared__ float red[WAVES_PER_BLOCK][32][9];

    const int lane = threadIdx.x & 31;
    const int wave = threadIdx.x >> 5;
    const int n0   = blockIdx.x * 16;        // output-feature tile base

    // WMMA f32 16x16x4 operand layout (ISA 7.12.2):
    //   A (16x4): lane&15 = M (batch), lane>>4 selects K pair {0,1} vs {2,3}
    //   B (4x16): lane&15 = N (out col), lane>>4 selects K pair, VGPR = K lsb
    const int n  = lane & 15;                // N within tile / M for A
    const int kb = (lane >> 4) << 1;         // 0 or 2: base K of this lane's pair
    const int m  = n;                        // batch row this lane feeds into A
    const bool mval = (m < BATCH);           // rows 8..15 are zero padding

    const int k0 = wave * KSLICE;

    const size_t wofs = (size_t)(n0 + n) * IN_F + (size_t)(k0 + kb);
    const float* po = outliers + wofs;
    const float* pu = updates  + wofs;
    const float* pc = cweight  + wofs;
    const float* px = x + (size_t)(mval ? m : 0) * IN_F + (size_t)(k0 + kb);

    v8f c = {0.f, 0.f, 0.f, 0.f, 0.f, 0.f, 0.f, 0.f};

    #pragma unroll 4
    for (int i = 0; i < KSLICE; i += 4) {
        // A fragment: x[batch][k], zero for padded batches
        v2f a = *(const v2f*)(px + i);
        if (!mval) { a.x = 0.f; a.y = 0.f; }

        // B fragment: reconstruct weight on the fly; weights are stream-once
        // -> non-temporal so they don't evict x / bias from L2.
        v2f o2 = __builtin_nontemporal_load((const v2f*)(po + i));
        v2f u2 = __builtin_nontemporal_load((const v2f*)(pu + i));
        v2f w2 = __builtin_nontemporal_load((const v2f*)(pc + i));
        v2f b2;
        b2.x = __builtin_fmaf(o2.x, u2.x, w2.x);
        b2.y = __builtin_fmaf(o2.y, u2.y, w2.y);

        // D(16x16 f32) += A(16x4) * B(4x16)
        c = __builtin_amdgcn_wmma_f32_16x16x4_f32(false, a, false, b2,
                                                  (short)0, c, false, false);
    }

    // Cross-wave K reduction through LDS.
    #pragma unroll
    for (int r = 0; r < 8; ++r) red[wave][lane][r] = c[r];
    __syncthreads();

    if (wave == 0) {
        float acc[8];
        #pragma unroll
        for (int r = 0; r < 8; ++r) acc[r] = 0.f;
        #pragma unroll
        for (int w = 0; w < WAVES_PER_BLOCK; ++w) {
            #pragma unroll
            for (int r = 0; r < 8; ++r) acc[r] += red[w][lane][r];
        }
        // C/D layout: lanes 0..15 hold M=0..7 in VGPRs 0..7 -> batch r, col lane
        if (lane < 16) {
            const int o = n0 + lane;
            const float bv = bias[o];
            #pragma unroll
            for (int r = 0; r < 8; ++r)
                out[(size_t)r * OUT_F + o] = acc[r] + bv;
        }
    }
}

extern "C" void kernel_launch(void* const* d_in, const int* in_sizes, int n_in,
                              void* d_out, int out_size, void* d_ws, size_t ws_size,
                              hipStream_t stream) {
    const float* x    = (const float*)d_in[0];  // [8, 4096]
    const float* outl = (const float*)d_in[1];  // [11008, 4096]
    const float* upd  = (const float*)d_in[2];  // [11008, 4096]
    const float* cw   = (const float*)d_in[3];  // [11008, 4096]
    const float* bias = (const float*)d_in[4];  // [11008]
    float* out        = (float*)d_out;          // [8, 11008]

    dim3 grid(OUT_F / 16);   // 688 output tiles
    dim3 block(256);         // 8 waves of 32
    OutlierQuantLinear_wmma_kernel<<<grid, block, 0, stream>>>(
        x, outl, upd, cw, bias, out);
}